// GATGTParallel_67654324847549
// MI455X (gfx1250) — compile-verified
//
#include <hip/hip_runtime.h>
#include <math.h>

typedef float v2f __attribute__((ext_vector_type(2)));
typedef float v8f __attribute__((ext_vector_type(8)));

#define NN 20000
#define EE 320000
#define DIN 128
#define HH 4
#define HC 256   // H*H1
#define BB 4096
#define DECC 512
#define EG_GAT (EE + NN)

// ---------------------------------------------------------------------------
// utility kernels
// ---------------------------------------------------------------------------
__global__ void fill_f32(float* __restrict__ p, float v, int n) {
  int i = blockIdx.x * blockDim.x + threadIdx.x;
  if (i < n) p[i] = v;
}

__device__ inline void atomicMaxF(float* addr, float val) {
  int* ia = reinterpret_cast<int*>(addr);
  int old = *ia;
  while (true) {
    float cur = __int_as_float(old);
    if (cur >= val) return;
    int assumed = old;
    old = atomicCAS(ia, assumed, __float_as_int(val));
    if (old == assumed) return;
  }
}

// ---------------------------------------------------------------------------
// LDS-staged WMMA fp32 GEMM: C[rows,M] = act(A[rows,K] @ W[K,M] + bias)
// block = 128 threads = 4 waves; block tile = 32x32 C (one 16x16 per wave);
// K tiled by 32, staged through LDS with coalesced float4 loads.
// A tile padded to stride 36 words -> conflict-free ds_load_b64 fragments.
// rows % 32 == 0, M % 32 == 0, K % 32 == 0 (all shapes here satisfy this).
// ---------------------------------------------------------------------------
__global__ __launch_bounds__(128) void wmma_gemm_f32(
    const float* __restrict__ A, int lda,
    const float* __restrict__ W, int ldb,
    const float* __restrict__ bias,
    float* __restrict__ C, int ldc,
    int K, int act) {
  __shared__ float aT[32 * 36];
  __shared__ float bT[32 * 32];

  const int t = threadIdx.x;
  const int wave = t >> 5;
  const int lane = t & 31;
  const int r = lane & 15;
  const int kk = (lane >> 4) * 2;         // A/B fragment K sub-offset
  const int wr = (wave >> 1) * 16;        // wave's row half within block tile
  const int wc = (wave & 1) * 16;         // wave's col half within block tile
  const int row0 = blockIdx.y * 32;
  const int col0 = blockIdx.x * 32;

  v8f acc;
  for (int i = 0; i < 8; ++i) acc[i] = 0.0f;

  for (int k0 = 0; k0 < K; k0 += 32) {
    // stage A (32 rows x 32 k) and B (32 k x 32 cols): 256 float4 each,
    // 2 per thread, fully coalesced.
#pragma unroll
    for (int i = 0; i < 2; ++i) {
      int fi = t + i * 128;
      int rr = fi >> 3;
      int c4 = (fi & 7) << 2;
      float4 av = *(const float4*)(A + (size_t)(row0 + rr) * lda + k0 + c4);
      *(float4*)(&aT[rr * 36 + c4]) = av;
      float4 bv = *(const float4*)(W + (size_t)(k0 + rr) * ldb + col0 + c4);
      *(float4*)(&bT[rr * 32 + c4]) = bv;
    }
    __syncthreads();

    if (k0 + 32 < K) {  // prefetch next K-slab (uniform branch, EXEC stays full)
      __builtin_prefetch(A + (size_t)(row0 + (t >> 3)) * lda + k0 + 32, 0, 3);
      __builtin_prefetch(W + (size_t)(k0 + 32 + (t >> 3)) * ldb + col0, 0, 3);
    }

#pragma unroll
    for (int kk0 = 0; kk0 < 32; kk0 += 4) {
      v2f a, b;
      a.x = aT[(wr + r) * 36 + kk0 + kk];
      a.y = aT[(wr + r) * 36 + kk0 + kk + 1];
      b.x = bT[(kk0 + kk) * 32 + wc + r];
      b.y = bT[(kk0 + kk + 1) * 32 + wc + r];
      acc = __builtin_amdgcn_wmma_f32_16x16x4_f32(false, a, false, b,
                                                  (short)0, acc, false, false);
    }
    __syncthreads();
  }

  const int rowoff = (lane >> 4) * 8;
#pragma unroll
  for (int i = 0; i < 8; ++i) {
    int row = row0 + wr + rowoff + i;
    int col = col0 + wc + r;
    float v = acc[i];
    if (bias) v += bias[col];
    if (act == 1) v = fmaxf(v, 0.0f);
    C[(size_t)row * ldc + col] = v;
  }
}

// ---------------------------------------------------------------------------
// GAT attention scalars: alS[n,h] = sum_c h[n,h,c]*a_src[h,c]; same for alD
// ---------------------------------------------------------------------------
__global__ void al_kernel(const float* __restrict__ h,
                          const float* __restrict__ asrc,
                          const float* __restrict__ adst,
                          int n, int Hh, int C,
                          float* __restrict__ alS, float* __restrict__ alD) {
  int i = blockIdx.x * blockDim.x + threadIdx.x;
  if (i >= n * Hh) return;
  int node = i / Hh, hd = i % Hh;
  const float* hp = h + (size_t)node * Hh * C + (size_t)hd * C;
  const float* as = asrc + hd * C;
  const float* ad = adst + hd * C;
  float s = 0.f, d = 0.f;
  for (int c = 0; c < C; c += 4) {
    float4 hv = *(const float4*)(hp + c);
    float4 av = *(const float4*)(as + c);
    float4 dv = *(const float4*)(ad + c);
    s += hv.x * av.x + hv.y * av.y + hv.z * av.z + hv.w * av.w;
    d += hv.x * dv.x + hv.y * dv.y + hv.z * dv.z + hv.w * dv.w;
  }
  alS[i] = s; alD[i] = d;
}

// GAT edge logits (leaky_relu 0.2) + running segment max. Edges >= E are
// self loops (node = eg - E).
__global__ void gat_edge_max(const float* __restrict__ alS,
                             const float* __restrict__ alD,
                             const int* __restrict__ src,
                             const int* __restrict__ dst,
                             int E, int EG, int Hh,
                             float* __restrict__ eb, float* __restrict__ segm) {
  int idx = blockIdx.x * blockDim.x + threadIdx.x;
  if (idx >= EG * Hh) return;
  int eg = idx / Hh, hd = idx % Hh;
  int s = (eg < E) ? src[eg] : (eg - E);
  int d = (eg < E) ? dst[eg] : (eg - E);
  float v = alS[s * Hh + hd] + alD[d * Hh + hd];
  v = (v > 0.f) ? v : 0.2f * v;
  eb[idx] = v;
  atomicMaxF(&segm[d * Hh + hd], v);
}

// TransformerConv edge logits: e = dot(q[dst],k[src])/sqrt(C) + segment max
__global__ void gt_edge_dot(const float* __restrict__ q,
                            const float* __restrict__ k,
                            const int* __restrict__ src,
                            const int* __restrict__ dst,
                            int E, int Hh, int C, float scale,
                            float* __restrict__ eb, float* __restrict__ segm) {
  int idx = blockIdx.x * blockDim.x + threadIdx.x;
  if (idx >= E * Hh) return;
  int e = idx / Hh, hd = idx % Hh;
  int s = src[e], d = dst[e];
  const float* qp = q + (size_t)d * Hh * C + (size_t)hd * C;
  const float* kp = k + (size_t)s * Hh * C + (size_t)hd * C;
  float acc = 0.f;
  for (int c = 0; c < C; c += 4) {
    float4 qv = *(const float4*)(qp + c);
    float4 kv = *(const float4*)(kp + c);
    acc += qv.x * kv.x + qv.y * kv.y + qv.z * kv.z + qv.w * kv.w;
  }
  float v = acc * scale;
  eb[idx] = v;
  atomicMaxF(&segm[d * Hh + hd], v);
}

// p = exp(e - max); segment sum of p
__global__ void edge_expsum(float* __restrict__ eb,
                            const int* __restrict__ dst,
                            int E, int EG, int Hh,
                            const float* __restrict__ segm,
                            float* __restrict__ segs) {
  int idx = blockIdx.x * blockDim.x + threadIdx.x;
  if (idx >= EG * Hh) return;
  int eg = idx / Hh, hd = idx % Hh;
  int d = (eg < E) ? dst[eg] : (eg - E);
  float p = expf(eb[idx] - segm[d * Hh + hd]);
  eb[idx] = p;
  atomicAdd(&segs[d * Hh + hd], p);
}

// acc[dst,h,c:c+4] += (p / (sum + 1e-16)) * feat[src,h,c:c+4]
// one float4 of channels per thread: 1 b128 gather + 4 f32 atomics.
__global__ void edge_agg4(const float* __restrict__ eb,
                          const float* __restrict__ segs,
                          const float* __restrict__ feat,
                          const int* __restrict__ src,
                          const int* __restrict__ dst,
                          int E, int EG, int Hh, int C4,  // C4 = (Hh*C)/4 per edge
                          float* __restrict__ acc) {
  int idx = blockIdx.x * blockDim.x + threadIdx.x;
  if (idx >= EG * C4) return;
  int eg = idx / C4;
  int q4 = idx % C4;                 // float4 index within the Hh*C features
  int C = (C4 * 4) / Hh;
  int hd = (q4 * 4) / C;
  int s = (eg < E) ? src[eg] : (eg - E);
  int d = (eg < E) ? dst[eg] : (eg - E);
  float alpha = eb[eg * Hh + hd] / (segs[d * Hh + hd] + 1e-16f);
  float4 f = *(const float4*)(feat + (size_t)s * Hh * C + q4 * 4);
  float* ap = acc + (size_t)d * Hh * C + q4 * 4;
  atomicAdd(ap + 0, alpha * f.x);
  atomicAdd(ap + 1, alpha * f.y);
  atomicAdd(ap + 2, alpha * f.z);
  atomicAdd(ap + 3, alpha * f.w);
}

// out[n, j] = prelu(acc[n,j] + bias[j], a[j])   (out has stride ldout)
__global__ void bias_prelu(const float* __restrict__ acc,
                           const float* __restrict__ bias,
                           const float* __restrict__ pa,
                           float* __restrict__ out, int ldout,
                           int n, int M) {
  int idx = blockIdx.x * blockDim.x + threadIdx.x;
  if (idx >= n * M) return;
  int r = idx / M, j = idx % M;
  float v = acc[idx] + bias[j];
  out[(size_t)r * ldout + j] = (v >= 0.f) ? v : pa[j] * v;
}

// out[n, j] = prelu(acc[n,j] + skip[n,j], a[j])
__global__ void skip_prelu(const float* __restrict__ acc,
                           const float* __restrict__ skip,
                           const float* __restrict__ pa,
                           float* __restrict__ out, int ldout,
                           int n, int M) {
  int idx = blockIdx.x * blockDim.x + threadIdx.x;
  if (idx >= n * M) return;
  int r = idx / M, j = idx % M;
  float v = acc[idx] + skip[idx];
  out[(size_t)r * ldout + j] = (v >= 0.f) ? v : pa[j] * v;
}

// column sums of x[n,32] into out[32] (out must be pre-zeroed)
__global__ void colsum_kernel(const float* __restrict__ x, int n,
                              float* __restrict__ out) {
  __shared__ float sh[32];
  if (threadIdx.x < 32) sh[threadIdx.x] = 0.f;
  __syncthreads();
  int total = n * 32;
  for (int i = blockIdx.x * blockDim.x + threadIdx.x; i < total;
       i += gridDim.x * blockDim.x)
    atomicAdd(&sh[i & 31], x[i]);
  __syncthreads();
  if (threadIdx.x < 32) atomicAdd(&out[threadIdx.x], sh[threadIdx.x]);
}

// h = sigmoid(mean) @ mlp1_w + mlp1_b   (for both graphs), 1 block / 32 thr
__global__ void summary_kernel(const float* __restrict__ cso,
                               const float* __restrict__ csa,
                               const float* __restrict__ w,
                               const float* __restrict__ b, float invN,
                               float* __restrict__ ho, float* __restrict__ ha) {
  __shared__ float so[32], sa[32];
  int j = threadIdx.x;
  so[j] = 1.f / (1.f + expf(-cso[j] * invN));
  sa[j] = 1.f / (1.f + expf(-csa[j] * invN));
  __syncthreads();
  float ao = b[j], aa = b[j];
  for (int i = 0; i < 32; ++i) { ao += so[i] * w[i * 32 + j]; aa += sa[i] * w[i * 32 + j]; }
  ho[j] = ao; ha[j] = aa;
}

// t[i] = sum_j disc_w[i,j] * h[j]  for both summaries
__global__ void disc_prep(const float* __restrict__ w,
                          const float* __restrict__ ho,
                          const float* __restrict__ ha,
                          float* __restrict__ to, float* __restrict__ ta) {
  int i = threadIdx.x;
  float so = 0.f, sa = 0.f;
  for (int j = 0; j < 32; ++j) { so += w[i * 32 + j] * ho[j]; sa += w[i * 32 + j] * ha[j]; }
  to[i] = so; ta[i] = sa;
}

// advrow[j] = sum_m adv_w[j,m];  advrow[32] = sum(adv_b)
__global__ void adv_prep(const float* __restrict__ w,
                         const float* __restrict__ b,
                         float* __restrict__ adv) {
  int j = threadIdx.x;
  if (j < 32) {
    float s = 0.f;
    for (int m = 0; m < 32; ++m) s += w[j * 32 + m];
    adv[j] = s;
  } else if (j == 32) {
    float s = 0.f;
    for (int m = 0; m < 32; ++m) s += b[m];
    adv[32] = s;
  }
}

// per-node bilinear discriminator scores + adversarial logits
__global__ void node_out_kernel(const float* __restrict__ x2o,
                                const float* __restrict__ x2a,
                                const float* __restrict__ to,
                                const float* __restrict__ ta,
                                const float* __restrict__ discb,
                                const float* __restrict__ adv,
                                float* __restrict__ ret_os,
                                float* __restrict__ ret_osa,
                                float* __restrict__ logits) {
  int n = blockIdx.x * blockDim.x + threadIdx.x;
  if (n >= NN) return;
  const float* po = x2o + (size_t)n * 32;
  const float* pa = x2a + (size_t)n * 32;
  float doo = 0.f, dao = 0.f, daa = 0.f, doa = 0.f, lo = 0.f, la = 0.f;
  for (int j = 0; j < 32; ++j) {
    float vo = po[j], va = pa[j];
    doo += vo * to[j]; dao += va * to[j];
    daa += va * ta[j]; doa += vo * ta[j];
    lo += vo * adv[j]; la += va * adv[j];
  }
  float db = discb[0], sb = adv[32];
  ret_os[2 * n] = doo + db;  ret_os[2 * n + 1] = dao + db;
  ret_osa[2 * n] = daa + db; ret_osa[2 * n + 1] = doa + db;
  logits[n] = lo + sb; logits[NN + n] = la + sb;
}

// hcat[b, 0:32] = x2_o[idx0[b]];  hcat[b, 32:64] = x2_o[idx1[b]]
__global__ void gather_kernel(const float* __restrict__ x2o,
                              const int* __restrict__ i0,
                              const int* __restrict__ i1,
                              float* __restrict__ hcat) {
  int idx = blockIdx.x * blockDim.x + threadIdx.x;
  if (idx >= BB * 16) return;            // one float4 per thread
  int b = idx >> 4, j4 = idx & 15;
  int node = (j4 < 8) ? i0[b] : i1[b];
  float4 v = *(const float4*)(x2o + (size_t)node * 32 + (j4 & 7) * 4);
  *(float4*)(hcat + (size_t)b * 64 + j4 * 4) = v;
}

// decoder heads: log = hh @ fus_w2 + b2 ; log1 = hh @ fus_w3 + b3
__global__ void head_kernel(const float* __restrict__ hh,
                            const float* __restrict__ w2,
                            const float* __restrict__ b2,
                            const float* __restrict__ w3,
                            const float* __restrict__ b3,
                            float* __restrict__ logo,
                            float* __restrict__ log1o) {
  int b = blockIdx.x * blockDim.x + threadIdx.x;
  if (b >= BB) return;
  const float* hv = hh + (size_t)b * DECC;
  float a0 = 0.f, a1 = 0.f, c0 = 0.f, c1 = 0.f;
  for (int i = 0; i < DECC; ++i) {
    float h = hv[i];
    a0 += h * w2[i * 2 + 0]; a1 += h * w2[i * 2 + 1];
    c0 += h * w3[i * 2 + 0]; c1 += h * w3[i * 2 + 1];
  }
  logo[2 * b] = a0 + b2[0]; logo[2 * b + 1] = a1 + b2[1];
  log1o[2 * b] = c0 + b3[0]; log1o[2 * b + 1] = c1 + b3[1];
}

// ---------------------------------------------------------------------------
// host orchestration
// ---------------------------------------------------------------------------
static inline dim3 g1(int n) { return dim3((n + 255) / 256); }

extern "C" void kernel_launch(void* const* d_in, const int* in_sizes, int n_in,
                              void* d_out, int out_size, void* d_ws, size_t ws_size,
                              hipStream_t stream) {
  const float* x_o = (const float*)d_in[0];
  const float* x_a = (const float*)d_in[1];
  const float* gat1_w = (const float*)d_in[2];
  const float* gat1_asrc = (const float*)d_in[3];
  const float* gat1_adst = (const float*)d_in[4];
  const float* gat1_b = (const float*)d_in[5];
  const float* gat2_w = (const float*)d_in[6];
  const float* gat2_asrc = (const float*)d_in[7];
  const float* gat2_adst = (const float*)d_in[8];
  const float* gat2_b = (const float*)d_in[9];
  const float* pg1 = (const float*)d_in[10];
  const float* pg2 = (const float*)d_in[11];
  const float* gt1_wq = (const float*)d_in[12];
  const float* gt1_bq = (const float*)d_in[13];
  const float* gt1_wk = (const float*)d_in[14];
  const float* gt1_bk = (const float*)d_in[15];
  const float* gt1_wv = (const float*)d_in[16];
  const float* gt1_bv = (const float*)d_in[17];
  const float* gt1_ws = (const float*)d_in[18];
  const float* gt1_bs = (const float*)d_in[19];
  const float* gt2_wq = (const float*)d_in[20];
  const float* gt2_bq = (const float*)d_in[21];
  const float* gt2_wk = (const float*)d_in[22];
  const float* gt2_bk = (const float*)d_in[23];
  const float* gt2_wv = (const float*)d_in[24];
  const float* gt2_bv = (const float*)d_in[25];
  const float* gt2_ws = (const float*)d_in[26];
  const float* gt2_bs = (const float*)d_in[27];
  const float* pt1 = (const float*)d_in[28];
  const float* pt2 = (const float*)d_in[29];
  const float* fuse_w = (const float*)d_in[30];
  const float* fuse_b = (const float*)d_in[31];
  const float* mlp1_w = (const float*)d_in[32];
  const float* mlp1_b = (const float*)d_in[33];
  const float* disc_w = (const float*)d_in[34];
  const float* disc_b = (const float*)d_in[35];
  const float* fus_w1 = (const float*)d_in[36];
  const float* fus_b1 = (const float*)d_in[37];
  const float* fus_w2 = (const float*)d_in[38];
  const float* fus_b2 = (const float*)d_in[39];
  const float* fus_w3 = (const float*)d_in[40];
  const float* fus_b3 = (const float*)d_in[41];
  const float* adv_w = (const float*)d_in[42];
  const float* adv_b = (const float*)d_in[43];
  const int* edge_index = (const int*)d_in[44];
  const int* idxp = (const int*)d_in[45];
  const int* src = edge_index;
  const int* dst = edge_index + EE;
  const int* idx0 = idxp;
  const int* idx1 = idxp + BB;

  // output layout (flat tuple order)
  float* out = (float*)d_out;
  float* o_log = out;                       // [B,2]
  float* o_ret = out + 8192;                // [N,2]
  float* o_reta = out + 48192;              // [N,2]
  float* o_x2o = out + 88192;               // [N,32]
  float* o_logits = out + 728192;           // [1,2N]
  float* o_log1 = out + 768192;             // [B,2]

  // workspace carve-up (floats)
  float* w = (float*)d_ws;
  size_t off = 0;
  float* bufH = w + off; off += (size_t)NN * HC;
  float* bufQ = w + off; off += (size_t)NN * HC;
  float* bufK = w + off; off += (size_t)NN * HC;
  float* bufV = w + off; off += (size_t)NN * HC;
  float* bufS = w + off; off += (size_t)NN * HC;
  float* accB = w + off; off += (size_t)NN * HC;
  float* edgeE = w + off; off += (size_t)EG_GAT * HH;
  float* alS = w + off; off += (size_t)NN * HH;
  float* alD = w + off; off += (size_t)NN * HH;
  float* segm = w + off; off += (size_t)NN * HH;
  float* segs = w + off; off += (size_t)NN * HH;
  float* cg_o = w + off; off += (size_t)NN * 64;
  float* cg_a = w + off; off += (size_t)NN * 64;
  float* x2a = w + off; off += (size_t)NN * 32;
  float* hcat = w + off; off += (size_t)BB * 64;
  float* hhb = w + off; off += (size_t)BB * DECC;
  float* cs_o = w + off; off += 32;
  float* cs_a = w + off; off += 32;
  float* h_os = w + off; off += 32;
  float* h_osa = w + off; off += 32;
  float* t_os = w + off; off += 32;
  float* t_osa = w + off; off += 32;
  float* advrow = w + off; off += 64;

  const float NEGINF = -3.4e38f;
  auto gemm = [&](const float* A, int lda, const float* W_, int ldb,
                  const float* bias, float* C, int ldc, int rows, int K,
                  int M, int act) {
    dim3 grid(M / 32, rows / 32), blk(128);
    wmma_gemm_f32<<<grid, blk, 0, stream>>>(A, lda, W_, ldb, bias, C, ldc, K, act);
  };

  auto run_graph = [&](const float* x, float* cg, float* x2out) {
    // ---------------- GAT layer 1: H=4, C=64, with self loops -------------
    gemm(x, DIN, gat1_w, HC, nullptr, bufH, HC, NN, DIN, HC, 0);
    al_kernel<<<g1(NN * HH), 256, 0, stream>>>(bufH, gat1_asrc, gat1_adst, NN, HH, 64, alS, alD);
    fill_f32<<<g1(NN * HH), 256, 0, stream>>>(segm, NEGINF, NN * HH);
    fill_f32<<<g1(NN * HH), 256, 0, stream>>>(segs, 0.f, NN * HH);
    fill_f32<<<g1(NN * HC), 256, 0, stream>>>(accB, 0.f, NN * HC);
    gat_edge_max<<<g1(EG_GAT * HH), 256, 0, stream>>>(alS, alD, src, dst, EE, EG_GAT, HH, edgeE, segm);
    edge_expsum<<<g1(EG_GAT * HH), 256, 0, stream>>>(edgeE, dst, EE, EG_GAT, HH, segm, segs);
    edge_agg4<<<g1(EG_GAT * (HC / 4)), 256, 0, stream>>>(edgeE, segs, bufH, src, dst, EE, EG_GAT, HH, HC / 4, accB);
    bias_prelu<<<g1(NN * HC), 256, 0, stream>>>(accB, gat1_b, pg1, bufQ, HC, NN, HC);

    // ---------------- GAT layer 2: H=1, C=32, with self loops -------------
    gemm(bufQ, HC, gat2_w, 32, nullptr, bufK, 32, NN, HC, 32, 0);
    al_kernel<<<g1(NN), 256, 0, stream>>>(bufK, gat2_asrc, gat2_adst, NN, 1, 32, alS, alD);
    fill_f32<<<g1(NN), 256, 0, stream>>>(segm, NEGINF, NN);
    fill_f32<<<g1(NN), 256, 0, stream>>>(segs, 0.f, NN);
    fill_f32<<<g1(NN * 32), 256, 0, stream>>>(accB, 0.f, NN * 32);
    gat_edge_max<<<g1(EG_GAT), 256, 0, stream>>>(alS, alD, src, dst, EE, EG_GAT, 1, edgeE, segm);
    edge_expsum<<<g1(EG_GAT), 256, 0, stream>>>(edgeE, dst, EE, EG_GAT, 1, segm, segs);
    edge_agg4<<<g1(EG_GAT * 8), 256, 0, stream>>>(edgeE, segs, bufK, src, dst, EE, EG_GAT, 1, 8, accB);
    bias_prelu<<<g1(NN * 32), 256, 0, stream>>>(accB, gat2_b, pg2, cg, 64, NN, 32);

    // ---------------- GT layer 1: H=4, C=64, no self loops ----------------
    gemm(x, DIN, gt1_wq, HC, gt1_bq, bufQ, HC, NN, DIN, HC, 0);
    gemm(x, DIN, gt1_wk, HC, gt1_bk, bufK, HC, NN, DIN, HC, 0);
    gemm(x, DIN, gt1_wv, HC, gt1_bv, bufV, HC, NN, DIN, HC, 0);
    gemm(x, DIN, gt1_ws, HC, gt1_bs, bufS, HC, NN, DIN, HC, 0);
    fill_f32<<<g1(NN * HH), 256, 0, stream>>>(segm, NEGINF, NN * HH);
    fill_f32<<<g1(NN * HH), 256, 0, stream>>>(segs, 0.f, NN * HH);
    fill_f32<<<g1(NN * HC), 256, 0, stream>>>(accB, 0.f, NN * HC);
    gt_edge_dot<<<g1(EE * HH), 256, 0, stream>>>(bufQ, bufK, src, dst, EE, HH, 64, 0.125f, edgeE, segm);
    edge_expsum<<<g1(EE * HH), 256, 0, stream>>>(edgeE, dst, EE, EE, HH, segm, segs);
    edge_agg4<<<g1(EE * (HC / 4)), 256, 0, stream>>>(edgeE, segs, bufV, src, dst, EE, EE, HH, HC / 4, accB);
    skip_prelu<<<g1(NN * HC), 256, 0, stream>>>(accB, bufS, pt1, bufH, HC, NN, HC);

    // ---------------- GT layer 2: H=1, C=32, no self loops ----------------
    gemm(bufH, HC, gt2_wq, 32, gt2_bq, bufQ, 32, NN, HC, 32, 0);
    gemm(bufH, HC, gt2_wk, 32, gt2_bk, bufK, 32, NN, HC, 32, 0);
    gemm(bufH, HC, gt2_wv, 32, gt2_bv, bufV, 32, NN, HC, 32, 0);
    gemm(bufH, HC, gt2_ws, 32, gt2_bs, bufS, 32, NN, HC, 32, 0);
    fill_f32<<<g1(NN), 256, 0, stream>>>(segm, NEGINF, NN);
    fill_f32<<<g1(NN), 256, 0, stream>>>(segs, 0.f, NN);
    fill_f32<<<g1(NN * 32), 256, 0, stream>>>(accB, 0.f, NN * 32);
    gt_edge_dot<<<g1(EE), 256, 0, stream>>>(bufQ, bufK, src, dst, EE, 1, 32, 0.1767766953f, edgeE, segm);
    edge_expsum<<<g1(EE), 256, 0, stream>>>(edgeE, dst, EE, EE, 1, segm, segs);
    edge_agg4<<<g1(EE * 8), 256, 0, stream>>>(edgeE, segs, bufV, src, dst, EE, EE, 1, 8, accB);
    skip_prelu<<<g1(NN * 32), 256, 0, stream>>>(accB, bufS, pt2, cg + 32, 64, NN, 32);

    // ---------------- fuse ------------------------------------------------
    gemm(cg, 64, fuse_w, 32, fuse_b, x2out, 32, NN, 64, 32, 0);
  };

  run_graph(x_o, cg_o, o_x2o);
  run_graph(x_a, cg_a, x2a);

  // summaries
  fill_f32<<<1, 64, 0, stream>>>(cs_o, 0.f, 64);  // zeroes cs_o and cs_a
  colsum_kernel<<<256, 256, 0, stream>>>(o_x2o, NN, cs_o);
  colsum_kernel<<<256, 256, 0, stream>>>(x2a, NN, cs_a);
  summary_kernel<<<1, 32, 0, stream>>>(cs_o, cs_a, mlp1_w, mlp1_b, 1.0f / NN, h_os, h_osa);
  disc_prep<<<1, 32, 0, stream>>>(disc_w, h_os, h_osa, t_os, t_osa);
  adv_prep<<<1, 64, 0, stream>>>(adv_w, adv_b, advrow);
  node_out_kernel<<<g1(NN), 256, 0, stream>>>(o_x2o, x2a, t_os, t_osa, disc_b,
                                              advrow, o_ret, o_reta, o_logits);

  // decoder
  gather_kernel<<<g1(BB * 16), 256, 0, stream>>>(o_x2o, idx0, idx1, hcat);
  gemm(hcat, 64, fus_w1, DECC, fus_b1, hhb, DECC, BB, 64, DECC, 1 /*relu*/);
  head_kernel<<<g1(BB), 256, 0, stream>>>(hhb, fus_w2, fus_b2, fus_w3, fus_b3, o_log, o_log1);
}